// CBOW_38826504355945
// MI455X (gfx1250) — compile-verified
//
#include <hip/hip_runtime.h>

typedef __attribute__((ext_vector_type(2))) float v2f;
typedef __attribute__((ext_vector_type(4))) float v4f;
typedef __attribute__((ext_vector_type(8))) float v8f;

#define BATCH  16384
#define EMBED  300
#define NCTX   10
#define NNEG   20
#define NSCORE 21          // 1 center + 20 negatives
#define NCHUNK 75          // 300 floats = 75 float4 chunks

__device__ __forceinline__ float log_sigmoid(float x) {
    // stable: min(x,0) - log1p(exp(-|x|))
    return fminf(x, 0.0f) - log1pf(__expf(-fabsf(x)));
}

// One wave32 per batch element.
__global__ __launch_bounds__(32)
void cbow_loss_kernel(const int* __restrict__ context,
                      const int* __restrict__ center,
                      const int* __restrict__ negatives,
                      const float* __restrict__ ctx_w,
                      const float* __restrict__ cen_w,
                      float* __restrict__ partial) {
    __shared__ __align__(16) float m_lds[EMBED];   // ctx_mean
    __shared__ float s_lds[32];                    // 21 scores (+pad)

    const int b    = blockIdx.x;
    const int lane = threadIdx.x;

    // ---------- ctx_mean: mean of 10 gathered rows, float4-striped ----------
    v4f a0v = {0,0,0,0}, a1v = {0,0,0,0}, a2v = {0,0,0,0};
    const bool t2 = lane < (NCHUNK - 64);          // chunk 2 has 11 float4
    #pragma unroll
    for (int c = 0; c < NCTX; ++c) {
        const v4f* r4 = (const v4f*)(ctx_w + (long)context[b * NCTX + c] * EMBED);
        a0v += r4[lane];
        a1v += r4[lane + 32];
        if (t2) a2v += r4[lane + 64];
    }
    const float inv = 1.0f / (float)NCTX;
    a0v *= inv; a1v *= inv; a2v *= inv;
    ((v4f*)m_lds)[lane]      = a0v;
    ((v4f*)m_lds)[lane + 32] = a1v;
    if (t2) ((v4f*)m_lds)[lane + 64] = a2v;
    __syncthreads();

    // ---------- score rows: A-matrix rows for two 16x4 tiles ----------
    // tile0: j = 0..15 (j=0 center, j-1 = negatives 0..14)
    // tile1: j = 16..20 (negatives 15..19), rows 21..31 zero-padded
    const int  mrow = lane & 15;
    const int  half = lane >> 4;                   // 0 -> K{0,1}, 1 -> K{2,3}
    const long cidx = (long)center[b];
    long j0idx = (mrow == 0) ? cidx : (long)negatives[b * NNEG + (mrow - 1)];
    const float* rowp0 = cen_w + j0idx * EMBED;
    const bool act1 = (mrow < 5);
    long j1idx = act1 ? (long)negatives[b * NNEG + (15 + mrow)] : cidx;
    const float* rowp1 = cen_w + j1idx * EMBED;
    const float mask1 = act1 ? 1.0f : 0.0f;

    // ---------- K loop: D += A(16x4) x B(4x16); B = m[k..k+3] bcast along N ----
    v8f d0 = {0,0,0,0,0,0,0,0};
    v8f d1 = {0,0,0,0,0,0,0,0};
    const int koff = 2 * half;
    for (int k = 0; k < EMBED; k += 4) {
        v2f bfrag;
        bfrag.x = m_lds[k + koff];
        bfrag.y = m_lds[k + koff + 1];
        v2f a0;  a0.x = rowp0[k + koff];          a0.y = rowp0[k + koff + 1];
        v2f a1;  a1.x = rowp1[k + koff] * mask1;  a1.y = rowp1[k + koff + 1] * mask1;
        d0 = __builtin_amdgcn_wmma_f32_16x16x4_f32(false, a0, false, bfrag,
                                                   (short)0, d0, false, false);
        d1 = __builtin_amdgcn_wmma_f32_16x16x4_f32(false, a1, false, bfrag,
                                                   (short)0, d1, false, false);
    }

    // ---------- scatter scores (D row M is constant along N) ----------
    // VGPR r: lanes 0-15 hold M=r, lanes 16-31 hold M=r+8
    if (lane == 0) {
        #pragma unroll
        for (int r = 0; r < 8; ++r) s_lds[r] = d0[r];
        #pragma unroll
        for (int r = 0; r < 5; ++r) s_lds[16 + r] = d1[r];
    }
    if (lane == 16) {
        #pragma unroll
        for (int r = 0; r < 8; ++r) s_lds[8 + r] = d0[r];
    }
    __syncthreads();

    // ---------- per-element loss ----------
    float contrib = 0.0f;
    if (lane < NSCORE) {
        float sc = s_lds[lane];
        contrib = log_sigmoid(lane == 0 ? sc : -sc);
    }
    #pragma unroll
    for (int off = 16; off >= 1; off >>= 1)
        contrib += __shfl_down(contrib, off, 32);
    if (lane == 0) partial[b] = contrib;
}

// Deterministic single-block tree reduction of 16384 partials.
__global__ __launch_bounds__(256)
void cbow_reduce_kernel(const float* __restrict__ partial, float* __restrict__ out) {
    __shared__ float sm[256];
    float s = 0.0f;
    for (int i = threadIdx.x; i < BATCH; i += 256) s += partial[i];
    sm[threadIdx.x] = s;
    __syncthreads();
    for (int off = 128; off > 0; off >>= 1) {
        if ((int)threadIdx.x < off) sm[threadIdx.x] += sm[threadIdx.x + off];
        __syncthreads();
    }
    if (threadIdx.x == 0) out[0] = -sm[0] * (1.0f / (float)BATCH);
}

extern "C" void kernel_launch(void* const* d_in, const int* in_sizes, int n_in,
                              void* d_out, int out_size, void* d_ws, size_t ws_size,
                              hipStream_t stream) {
    const int*   context   = (const int*)d_in[0];
    const int*   center    = (const int*)d_in[1];
    const int*   negatives = (const int*)d_in[2];
    const float* ctx_w     = (const float*)d_in[3];
    const float* cen_w     = (const float*)d_in[4];
    float* out     = (float*)d_out;
    float* partial = (float*)d_ws;          // 16384 floats = 64 KB

    cbow_loss_kernel<<<dim3(BATCH), dim3(32), 0, stream>>>(
        context, center, negatives, ctx_w, cen_w, partial);
    cbow_reduce_kernel<<<dim3(1), dim3(256), 0, stream>>>(partial, out);
}